// router_20091857011524
// MI455X (gfx1250) — compile-verified
//
#include <hip/hip_runtime.h>
#include <math.h>

typedef __attribute__((ext_vector_type(2))) float v2f;
typedef __attribute__((ext_vector_type(8))) float v8f;

#define BB 8
#define TT 4096
#define DD 512
#define KK 4
#define NTOK (BB * TT)            // 32768 tokens
#define WAVES_PER_WG 8
#define TOK_PER_WAVE 16           // one 16-token WMMA tile per wave
#define NTILE (NTOK / TOK_PER_WAVE)   // 2048
#define NBLOCK (NTILE / WAVES_PER_WG) // 256

__global__ __launch_bounds__(256) void router_fused_kernel(
    const float* __restrict__ m0, const float* __restrict__ m1,
    const float* __restrict__ m2, const float* __restrict__ m3,
    const float* __restrict__ Wt, const float* __restrict__ bt,
    const float* __restrict__ Ws, const float* __restrict__ bs,
    const float* __restrict__ alpha, float* __restrict__ out)
{
    // De-interleaved router weights: wlds[k][d][j] = W[j][4*d+k]
    //   j in 0..3 -> W_top rows, j in 4..7 -> W_soft rows.  64 KB.
    __shared__ float wlds[KK * DD * 8];
    __shared__ float clds[WAVES_PER_WG][16][8];   // per-wave logits (M x N)
    __shared__ float wrt [WAVES_PER_WG][16][4];   // per-wave routing weights

    const int tid  = threadIdx.x;
    const int wave = tid >> 5;
    const int lane = tid & 31;

    // ---- Stage de-interleaved weights into LDS (once per workgroup) ----
    for (int idx = tid; idx < KK * DD * 8; idx += 256) {
        const int j = idx & 7;
        const int d = (idx >> 3) & (DD - 1);
        const int k = idx >> 12;
        const float v = (j < 4) ? Wt[j * (DD * KK) + 4 * d + k]
                                : Ws[(j - 4) * (DD * KK) + 4 * d + k];
        wlds[idx] = v;
    }
    __syncthreads();

    const int tile = blockIdx.x * WAVES_PER_WG + wave;  // 0..2047
    const int tok0 = tile * TOK_PER_WAVE;

    // WMMA fragment lane mapping (16x16x4 f32):
    //   A (16 tokens x 4 K): lane<16 holds A[lane][k0], A[lane][k0+1];
    //                        lane>=16 holds A[lane-16][k0+2], A[lane-16][k0+3]
    //   B (4 K x 16 cols):   lane<16 holds B[k0][lane], B[k0+1][lane]; etc.
    const int mrow = lane & 15;    // token within tile (A row / C row group)
    const int hi   = lane >> 4;    // K sub-pair select
    const int col  = lane & 7;     // weight column (cols 8..15 replicate 0..7)

    const float* mats[4] = { m0, m1, m2, m3 };

    v8f acc0 = {};
    v8f acc1 = {};

    #pragma unroll
    for (int k = 0; k < KK; ++k) {
        const float* abase = mats[k] + (size_t)(tok0 + mrow) * DD + 2 * hi;
        const float* wbase = &wlds[(k * DD) * 8 + col];
        #pragma unroll 4
        for (int d0 = 0; d0 < DD; d0 += 8) {
            // step 1: K = d0..d0+3
            v2f a0 = *(const v2f*)(abase + d0);
            v2f b0 = { wbase[(d0 + 2 * hi) * 8],
                       wbase[(d0 + 2 * hi + 1) * 8] };
            acc0 = __builtin_amdgcn_wmma_f32_16x16x4_f32(
                       false, a0, false, b0, (short)0, acc0, false, false);
            // step 2: K = d0+4..d0+7
            v2f a1 = *(const v2f*)(abase + d0 + 4);
            v2f b1 = { wbase[(d0 + 4 + 2 * hi) * 8],
                       wbase[(d0 + 4 + 2 * hi + 1) * 8] };
            acc1 = __builtin_amdgcn_wmma_f32_16x16x4_f32(
                       false, a1, false, b1, (short)0, acc1, false, false);
        }
    }
    v8f acc = acc0 + acc1;

    // ---- Spill useful logits (N < 8) to LDS: clds[wave][token][col] ----
    if ((lane & 15) < 8) {
        const int n = lane & 15;
        const int mb = hi ? 8 : 0;
        #pragma unroll
        for (int r = 0; r < 8; ++r)
            clds[wave][mb + r][n] = acc[r];
    }
    __syncthreads();

    // ---- Routing: one lane per token (lanes 0..15) ----
    if (lane < 16) {
        float lt[4], ls[4];
        #pragma unroll
        for (int j = 0; j < 4; ++j) {
            lt[j] = clds[wave][lane][j]     + bt[j];
            ls[j] = clds[wave][lane][4 + j] + bs[j];
        }
        // top-2 (ties -> lowest index, matching jax.lax.top_k)
        int i1 = 0; float v1 = lt[0];
        #pragma unroll
        for (int j = 1; j < 4; ++j)
            if (lt[j] > v1) { v1 = lt[j]; i1 = j; }
        int i2 = -1; float v2 = -3.4e38f;
        #pragma unroll
        for (int j = 0; j < 4; ++j)
            if (j != i1 && lt[j] > v2) { v2 = lt[j]; i2 = j; }
        const float e  = __expf(v2 - v1);        // v2 <= v1, stable
        const float p1 = 1.0f / (1.0f + e);
        const float p2 = 1.0f - p1;
        // full softmax over soft logits
        float mx = ls[0];
        #pragma unroll
        for (int j = 1; j < 4; ++j) mx = fmaxf(mx, ls[j]);
        float es[4], ssum = 0.0f;
        #pragma unroll
        for (int j = 0; j < 4; ++j) { es[j] = __expf(ls[j] - mx); ssum += es[j]; }
        const float rs = 1.0f / ssum;
        const float a  = 1.0f / (1.0f + __expf(-alpha[0]));
        #pragma unroll
        for (int kk2 = 0; kk2 < 4; ++kk2) {
            const float hard = (kk2 == i1) ? p1 : ((kk2 == i2) ? p2 : 0.0f);
            wrt[wave][lane][kk2] = a * hard + (1.0f - a) * (es[kk2] * rs);
        }
    }
    __syncthreads();

    // ---- Weighted sum: out[t][d] = sum_k w[t][k] * m_k[t][d]  (coalesced) ----
    for (int tok = 0; tok < TOK_PER_WAVE; ++tok) {
        const float w0 = wrt[wave][tok][0];
        const float w1 = wrt[wave][tok][1];
        const float w2 = wrt[wave][tok][2];
        const float w3 = wrt[wave][tok][3];
        const size_t row = (size_t)(tok0 + tok) * DD;
        #pragma unroll
        for (int it = 0; it < 4; ++it) {
            const int d = it * 128 + 4 * lane;
            float4 x0 = *(const float4*)(m0 + row + d);
            float4 x1 = *(const float4*)(m1 + row + d);
            float4 x2 = *(const float4*)(m2 + row + d);
            float4 x3 = *(const float4*)(m3 + row + d);
            float4 o;
            o.x = x0.x * w0 + x1.x * w1 + x2.x * w2 + x3.x * w3;
            o.y = x0.y * w0 + x1.y * w1 + x2.y * w2 + x3.y * w3;
            o.z = x0.z * w0 + x1.z * w1 + x2.z * w2 + x3.z * w3;
            o.w = x0.w * w0 + x1.w * w1 + x2.w * w2 + x3.w * w3;
            *(float4*)(out + row + d) = o;
        }
    }
}

extern "C" void kernel_launch(void* const* d_in, const int* in_sizes, int n_in,
                              void* d_out, int out_size, void* d_ws, size_t ws_size,
                              hipStream_t stream) {
    (void)in_sizes; (void)n_in; (void)d_ws; (void)ws_size; (void)out_size;
    const float* m0 = (const float*)d_in[0];
    const float* m1 = (const float*)d_in[1];
    const float* m2 = (const float*)d_in[2];
    const float* m3 = (const float*)d_in[3];
    const float* Wt = (const float*)d_in[4];
    const float* bt = (const float*)d_in[5];
    const float* Ws = (const float*)d_in[6];
    const float* bs = (const float*)d_in[7];
    const float* al = (const float*)d_in[8];
    float* out = (float*)d_out;

    router_fused_kernel<<<NBLOCK, 256, 0, stream>>>(
        m0, m1, m2, m3, Wt, bt, Ws, bs, al, out);
}